// MultiHeadLocationAwareAttention_18442589569761
// MI455X (gfx1250) — compile-verified
//
#include <hip/hip_runtime.h>
#include <math.h>

// Problem dims
static constexpr int Bn = 64, Tn = 2048, En = 512, Rn = 1024, An = 128;
static constexpr int Hn = 8, HDn = 16, NFn = 32, KSn = 31, PADn = 15;
static constexpr int CHUNK = 256;           // t-chunk per block in fused kernel
static constexpr int NCH = Tn / CHUNK;      // 8 chunks per batch
static constexpr float SCALE = 0.25f;       // 1/sqrt(16)
static constexpr float CPLX = 0.1f;

typedef float v2f __attribute__((ext_vector_type(2)));
typedef float v8f __attribute__((ext_vector_type(8)));

// ---------------- Kernel 1: Qh[b,a] = query[b]·Wq[a] + bq[a] ----------------
__global__ __launch_bounds__(128) void qh_kernel(
    const float* __restrict__ query, const float* __restrict__ Wq,
    const float* __restrict__ bq, float* __restrict__ Qh) {
  __shared__ float q[Rn];
  int b = blockIdx.x, tid = threadIdx.x;
  for (int i = tid; i < Rn; i += 128) q[i] = query[(size_t)b * Rn + i];
  __syncthreads();
  float acc = bq[tid];
  const float* w = Wq + (size_t)tid * Rn;
  for (int r = 0; r < Rn; ++r) acc += q[r] * w[r];
  Qh[b * An + tid] = acc;
}

// ---- Kernel 2: fold query into weights (P rows 8..15 zeroed for WMMA pad) --
__global__ __launch_bounds__(256) void proj_kernel(
    const float* __restrict__ Qh, const float* __restrict__ Wk,
    const float* __restrict__ bk, const float* __restrict__ Wl,
    float* __restrict__ P, float* __restrict__ Lc, float* __restrict__ qbk) {
  int m = blockIdx.x, b = blockIdx.y, tid = threadIdx.x;
  float* Prow = P + ((size_t)b * 16 + m) * En;
  if (m >= Hn) {
    Prow[tid] = 0.f;
    Prow[tid + 256] = 0.f;
    return;
  }
  __shared__ float qh[HDn];
  if (tid < HDn) qh[tid] = Qh[b * An + m * HDn + tid];
  __syncthreads();
  for (int half = 0; half < 2; ++half) {
    int e = tid + half * 256;
    float acc = 0.f;
#pragma unroll
    for (int d = 0; d < HDn; ++d) acc += qh[d] * Wk[(size_t)(m * HDn + d) * En + e];
    Prow[e] = SCALE * acc;
  }
  if (tid < NFn) {
    float acc = 0.f;
#pragma unroll
    for (int d = 0; d < HDn; ++d) acc += qh[d] * Wl[(m * HDn + d) * NFn + tid];
    Lc[(b * Hn + m) * NFn + tid] = CPLX * SCALE * acc;
  }
  if (tid == 0) {
    float acc = 0.f;
#pragma unroll
    for (int d = 0; d < HDn; ++d) acc += qh[d] * bk[m * HDn + d];
    qbk[b * Hn + m] = SCALE * acc;
  }
}

// ---- Kernel 3 (fused): per (b, 256-wide t-chunk):
//  phase 1: scores[8 x 256] via WMMA (P · mem^T), + conv/bias init, -> LDS + global
//  phase 2: chunk-local per-head max / exp / sum (one wave per head)
//  phase 3: context partials acc[8 x 512] = p~ · mem_chunk via WMMA (K = t)
static constexpr int PPITCH = 258;  // pLDS row pitch: 258 % 64 == 2 -> bank-conflict-free
__global__ __launch_bounds__(256) void scores_ctx_kernel(
    const float* __restrict__ P, const float* __restrict__ Lc,
    const float* __restrict__ qbk, const float* __restrict__ mem,
    const float* __restrict__ aw, const float* __restrict__ cw,
    const unsigned char* __restrict__ mask, float* __restrict__ scores,
    float* __restrict__ MZc, float* __restrict__ accPart) {
  __shared__ float cwLDS[NFn * 2 * KSn];           // conv weights (1984)
  __shared__ float awLDS[2 * (CHUNK + 2 * PADn)];  // padded attn-weight window
  __shared__ float LcLDS[Hn * NFn];
  __shared__ float qbkLDS[Hn];
  __shared__ float sLDS[Hn * CHUNK];               // raw scores (8 x 256)
  __shared__ float pLDS[16 * PPITCH];              // exp(s - Mc), rows 8..15 = 0

  int b = blockIdx.x;
  int chunk = blockIdx.y;
  int tchunk = chunk * CHUNK;
  int tid = threadIdx.x;

  // ---- phase 0: stage LDS ----
  for (int i = tid; i < NFn * 2 * KSn; i += 256) cwLDS[i] = cw[i];
  for (int i = tid; i < 2 * 286; i += 256) {
    int c = i / 286, j = i % 286;
    int t = tchunk - PADn + j;
    awLDS[i] = (t >= 0 && t < Tn) ? aw[((size_t)b * 2 + c) * Tn + t] : 0.f;
  }
  for (int i = tid; i < Hn * NFn; i += 256) LcLDS[i] = Lc[b * Hn * NFn + i];
  if (tid < Hn) qbkLDS[tid] = qbk[b * Hn + tid];
  for (int i = tid; i < 8 * PPITCH; i += 256) pLDS[8 * PPITCH + i] = 0.f;  // pad rows
  __syncthreads();

  int lane = tid & 31;
  int wid = tid >> 5;
  int n = lane & 15;
  int koff = (lane < 16) ? 0 : 2;
  const float* Arow = P + ((size_t)b * 16 + n) * En + koff;

  // ---- phase 1: score tiles (2 tiles of 16 t per wave) ----
  for (int tile = 0; tile < 2; ++tile) {
    int t0 = tchunk + wid * 32 + tile * 16;
    int t = t0 + n;
    v8f c;
#pragma unroll
    for (int v = 0; v < 8; ++v) c[v] = 0.f;
    if (lane < 16) {
      int dt = t - tchunk;
      float acc[8];
#pragma unroll
      for (int v = 0; v < 8; ++v) acc[v] = qbkLDS[v];
      for (int f = 0; f < NFn; ++f) {
        float lf = 0.f;
#pragma unroll
        for (int cc = 0; cc < 2; ++cc)
          for (int k = 0; k < KSn; ++k)
            lf += awLDS[cc * 286 + dt + k] * cwLDS[f * 62 + cc * KSn + k];
#pragma unroll
        for (int v = 0; v < 8; ++v) acc[v] += LcLDS[v * NFn + f] * lf;
      }
#pragma unroll
      for (int v = 0; v < 8; ++v) c[v] = acc[v];
    }
    const float* Brow = mem + ((size_t)b * Tn + t) * En + koff;
#pragma unroll 4
    for (int kk = 0; kk < En / 4; ++kk) {
      v2f af = *(const v2f*)(Arow + kk * 4);
      v2f bf = *(const v2f*)(Brow + kk * 4);
      c = __builtin_amdgcn_wmma_f32_16x16x4_f32(false, af, false, bf,
                                                (short)0, c, false, false);
    }
    if (lane < 16) {  // lanes 0-15 hold head rows 0-7
      bool mk = mask[(size_t)b * Tn + t] != 0;
      int dt = t - tchunk;
#pragma unroll
      for (int v = 0; v < 8; ++v) {
        float s = mk ? -INFINITY : c[v];
        scores[((size_t)b * Hn + v) * Tn + t] = s;
        sLDS[v * CHUNK + dt] = s;
      }
    }
  }
  __syncthreads();

  // ---- phase 2: chunk-local softmax numerators (wave wid <-> head wid) ----
  {
    int h = wid;  // 8 waves, 8 heads
    float m = -INFINITY;
    for (int i = lane; i < CHUNK; i += 32) m = fmaxf(m, sLDS[h * CHUNK + i]);
#pragma unroll
    for (int off = 16; off > 0; off >>= 1) m = fmaxf(m, __shfl_xor(m, off, 32));
    bool dead = (m == -INFINITY);
    float z = 0.f;
    for (int i = lane; i < CHUNK; i += 32) {
      float p = dead ? 0.f : expf(sLDS[h * CHUNK + i] - m);
      pLDS[h * PPITCH + i] = p;
      z += p;
    }
#pragma unroll
    for (int off = 16; off > 0; off >>= 1) z += __shfl_xor(z, off, 32);
    if (lane == 0) {
      size_t o = (((size_t)b * NCH + chunk) * Hn + h) * 2;
      MZc[o] = m;
      MZc[o + 1] = z;
    }
  }
  __syncthreads();

  // ---- phase 3: acc[h][e] = sum_t p~[h][t] * mem[t][e]  (WMMA, K = t) ----
  {
    v8f acc[4];
#pragma unroll
    for (int nt = 0; nt < 4; ++nt)
#pragma unroll
      for (int v = 0; v < 8; ++v) acc[nt][v] = 0.f;
    const float* Mb = mem + ((size_t)b * Tn + tchunk) * En;
    const float* Ap = pLDS + (lane & 15) * PPITCH + koff;
#pragma unroll 2
    for (int kk = 0; kk < CHUNK / 4; ++kk) {
      int r0 = 4 * kk + koff;  // t-local rows held by this half-wave
      v2f af = *(const v2f*)(Ap + 4 * kk);
#pragma unroll
      for (int nt = 0; nt < 4; ++nt) {
        int col = wid * 64 + nt * 16 + n;
        v2f bf;
        bf.x = Mb[(size_t)r0 * En + col];
        bf.y = Mb[(size_t)(r0 + 1) * En + col];
        acc[nt] = __builtin_amdgcn_wmma_f32_16x16x4_f32(false, af, false, bf,
                                                        (short)0, acc[nt],
                                                        false, false);
      }
    }
    if (lane < 16) {
      size_t base = ((size_t)b * NCH + chunk) * Hn;
#pragma unroll
      for (int nt = 0; nt < 4; ++nt) {
        int col = wid * 64 + nt * 16 + n;
#pragma unroll
        for (int v = 0; v < 8; ++v)
          accPart[(base + v) * En + col] = acc[nt][v];
      }
    }
  }
}

// ---- Kernel 4: combine chunk partials -> global (M,Z) per head + context ----
__global__ __launch_bounds__(256) void combine_kernel(
    const float* __restrict__ MZc, const float* __restrict__ accPart,
    float* __restrict__ MZ, float* __restrict__ ctx) {
  int b = blockIdx.x, tid = threadIdx.x;
  __shared__ float McS[NCH][Hn], ZcS[NCH][Hn], Fs[NCH][Hn];
  __shared__ float ZhS[Hn];
  if (tid < NCH * Hn) {
    int c = tid / Hn, h = tid % Hn;
    size_t o = (((size_t)b * NCH + c) * Hn + h) * 2;
    McS[c][h] = MZc[o];
    ZcS[c][h] = MZc[o + 1];
  }
  __syncthreads();
  if (tid < Hn) {
    int h = tid;
    float M = -INFINITY;
#pragma unroll
    for (int c = 0; c < NCH; ++c) M = fmaxf(M, McS[c][h]);
    float Z = 0.f;
#pragma unroll
    for (int c = 0; c < NCH; ++c) {
      float f = (McS[c][h] == -INFINITY) ? 0.f : expf(McS[c][h] - M);
      Fs[c][h] = f;
      Z += f * ZcS[c][h];
    }
    ZhS[h] = Z;
    MZ[(b * Hn + h) * 2] = M;
    MZ[(b * Hn + h) * 2 + 1] = Z;
  }
  __syncthreads();
  for (int e = tid; e < En; e += 256) {
    float acc = 0.f;
#pragma unroll
    for (int h = 0; h < Hn; ++h) {
      float s = 0.f;
#pragma unroll
      for (int c = 0; c < NCH; ++c)
        s += Fs[c][h] * accPart[(((size_t)b * NCH + c) * Hn + h) * En + e];
      acc += s / ZhS[h];
    }
    ctx[b * En + e] = acc * (1.0f / Hn);
  }
}

// ---- Kernel 5: final_weights[b,t] = mean_h exp(s - M_h)/Z_h ----
__global__ __launch_bounds__(256) void fw_kernel(
    const float* __restrict__ scores, const float* __restrict__ MZ,
    float* __restrict__ fw) {
  int idx = blockIdx.x * 256 + threadIdx.x;
  int b = idx >> 11;        // / Tn
  int t = idx & (Tn - 1);
  float acc = 0.f;
#pragma unroll
  for (int h = 0; h < Hn; ++h) {
    float M = MZ[(b * Hn + h) * 2];
    float Z = MZ[(b * Hn + h) * 2 + 1];
    acc += expf(scores[((size_t)b * Hn + h) * Tn + t] - M) / Z;
  }
  fw[idx] = acc * (1.0f / Hn);
}

extern "C" void kernel_launch(void* const* d_in, const int* in_sizes, int n_in,
                              void* d_out, int out_size, void* d_ws, size_t ws_size,
                              hipStream_t stream) {
  (void)in_sizes; (void)n_in; (void)out_size; (void)ws_size;
  const float* query  = (const float*)d_in[0];
  const float* memory = (const float*)d_in[1];
  const float* aw     = (const float*)d_in[3];
  const unsigned char* mask = (const unsigned char*)d_in[4];
  const float* Wq = (const float*)d_in[5];
  const float* bq = (const float*)d_in[6];
  const float* Wk = (const float*)d_in[7];
  const float* bk = (const float*)d_in[8];
  const float* conv_w = (const float*)d_in[13];
  const float* Wl = (const float*)d_in[14];

  float* out = (float*)d_out;
  float* ctx = out;               // [B,E]
  float* fw  = out + Bn * En;     // [B,T]

  float* ws  = (float*)d_ws;
  float* Qh  = ws;                                   // B*A            = 8192
  float* P   = Qh + Bn * An;                         // B*16*E         = 524288
  float* Lc  = P + (size_t)Bn * 16 * En;             // B*H*NF         = 16384
  float* qbk = Lc + Bn * Hn * NFn;                   // B*H            = 512
  float* scores = qbk + Bn * Hn;                     // B*H*T          = 1048576
  float* MZc = scores + (size_t)Bn * Hn * Tn;        // B*NCH*H*2      = 8192
  float* accPart = MZc + (size_t)Bn * NCH * Hn * 2;  // B*NCH*H*E      = 2097152
  float* MZ = accPart + (size_t)Bn * NCH * Hn * En;  // B*H*2          = 1024

  qh_kernel<<<dim3(Bn), dim3(128), 0, stream>>>(query, Wq, bq, Qh);
  proj_kernel<<<dim3(16, Bn), dim3(256), 0, stream>>>(Qh, Wk, bk, Wl, P, Lc, qbk);
  scores_ctx_kernel<<<dim3(Bn, NCH), dim3(256), 0, stream>>>(
      P, Lc, qbk, memory, aw, conv_w, mask, scores, MZc, accPart);
  combine_kernel<<<dim3(Bn), dim3(256), 0, stream>>>(MZc, accPart, MZ, ctx);
  fw_kernel<<<dim3(Bn * Tn / 256), dim3(256), 0, stream>>>(scores, MZ, fw);
}